// SDEModel_60120952209684
// MI455X (gfx1250) — compile-verified
//
#include <hip/hip_runtime.h>
#include <hip/hip_bf16.h>
#include <math.h>

#define NN   384
#define CTX  16
#define MB   8
#define GAM  64
#define G2   49
#define FNN  32

// output float offsets
#define O_DX    0
#define O_LOGDS 3072
#define O_LOGG  3080
#define O_H1N   3088
#define O_H2N   3600
#define O_SIG   3992

typedef float v2f __attribute__((ext_vector_type(2)));
typedef float v8f __attribute__((ext_vector_type(8)));

__device__ __forceinline__ float wave_sum(float v) {
    #pragma unroll
    for (int off = 16; off > 0; off >>= 1) v += __shfl_xor(v, off, 32);
    return v;
}
__device__ __forceinline__ float eluf(float v)  { return v > 0.0f ? v : __expf(v) - 1.0f; }
__device__ __forceinline__ float sigmf(float v) { return 1.0f / (1.0f + __expf(-v)); }

// ---------------------------------------------------------------------------
// Kernel A: stream W (226 MB) once; per slice k produce
//   Csum[k,i] = rowsum_i + colsum_i - 2*W[k,i,i] - W[k,i,k] - W[k,k,i]  (0 if i==k)
// Block = one k slice (576 KB), 384 threads = 12 waves, float4 coalesced.
// This kernel is the HBM-roofline limiter (~10us at 23.3 TB/s); everything
// else is noise.
// ---------------------------------------------------------------------------
__global__ __launch_bounds__(384)
void wreduce_kernel(const float* __restrict__ W, float* __restrict__ csum) {
    const int k    = blockIdx.x;
    const int tid  = threadIdx.x;
    const int wave = tid >> 5;
    const int lane = tid & 31;
    const float* __restrict__ Wk = W + (size_t)k * NN * NN;

    __shared__ float s_row[NN];
    __shared__ float s_colp[12][NN];

    float4 c0 = {0,0,0,0}, c1 = {0,0,0,0}, c2 = {0,0,0,0};
    #pragma unroll 1
    for (int r = 0; r < 32; ++r) {
        const int i = wave * 32 + r;
        const float4* __restrict__ row = (const float4*)(Wk + (size_t)i * NN);
        __builtin_prefetch(Wk + (size_t)(i + 2) * NN + lane * 4, 0, 1);
        float4 v0 = row[lane];
        float4 v1 = row[lane + 32];
        float4 v2 = row[lane + 64];
        c0.x += v0.x; c0.y += v0.y; c0.z += v0.z; c0.w += v0.w;
        c1.x += v1.x; c1.y += v1.y; c1.z += v1.z; c1.w += v1.w;
        c2.x += v2.x; c2.y += v2.y; c2.z += v2.z; c2.w += v2.w;
        float p = (v0.x + v0.y + v0.z + v0.w)
                + (v1.x + v1.y + v1.z + v1.w)
                + (v2.x + v2.y + v2.z + v2.w);
        p = wave_sum(p);
        if (lane == 0) s_row[i] = p;
    }
    {
        int b0 = 4 * lane;
        s_colp[wave][b0 + 0] = c0.x; s_colp[wave][b0 + 1] = c0.y;
        s_colp[wave][b0 + 2] = c0.z; s_colp[wave][b0 + 3] = c0.w;
        int b1 = 4 * (lane + 32);
        s_colp[wave][b1 + 0] = c1.x; s_colp[wave][b1 + 1] = c1.y;
        s_colp[wave][b1 + 2] = c1.z; s_colp[wave][b1 + 3] = c1.w;
        int b2 = 4 * (lane + 64);
        s_colp[wave][b2 + 0] = c2.x; s_colp[wave][b2 + 1] = c2.y;
        s_colp[wave][b2 + 2] = c2.z; s_colp[wave][b2 + 3] = c2.w;
    }
    __syncthreads();

    const int j = tid;                       // 384 threads == 384 columns
    float cs = 0.0f;
    #pragma unroll
    for (int w = 0; w < 12; ++w) cs += s_colp[w][j];
    float res = 0.0f;
    if (j != k) {
        res = s_row[j] + cs
            - 2.0f * Wk[(size_t)j * NN + j]
            - Wk[(size_t)j * NN + k]
            - Wk[(size_t)k * NN + j];
    }
    csum[(size_t)k * NN + j] = res;
}

// ---------------------------------------------------------------------------
// Kernel F: DiagonalFunc f_out[b,n].  Block = node n; thread = (b, h); one
// wave per batch row reduces the 32 hidden units.
// ---------------------------------------------------------------------------
__global__ __launch_bounds__(256)
void fdiag_kernel(const float* __restrict__ x, const float* __restrict__ t,
                  const float* __restrict__ ctx,
                  const float* __restrict__ fW1, const float* __restrict__ fb1,
                  const float* __restrict__ fW2, const float* __restrict__ fb2,
                  float* __restrict__ fout) {
    const int n = blockIdx.x;
    const int b = threadIdx.x >> 5;
    const int h = threadIdx.x & 31;
    const float* __restrict__ w = fW1 + (size_t)n * 18 * FNN;   // [d][h]
    const float tn = (t[b] - 40.0f) / 20.0f;

    float acc = fb1[n * FNN + h];
    acc += x[b * NN + n] * w[0 * FNN + h];
    acc += tn            * w[1 * FNN + h];
    #pragma unroll
    for (int c = 0; c < CTX; ++c)
        acc += ctx[b * CTX + c] * w[(2 + c) * FNN + h];

    float v = eluf(acc) * fW2[n * FNN + h];
    v = wave_sum(v);
    if (h == 0) fout[b * NN + n] = v + fb2[n];
}

// ---------------------------------------------------------------------------
// Kernel C: GRUs, hazard head, g-net.  Single block, 256 threads, staged LDS.
// ---------------------------------------------------------------------------
__global__ __launch_bounds__(256)
void small_kernel(const float* __restrict__ x,  const float* __restrict__ h1,
                  const float* __restrict__ h2, const float* __restrict__ t,
                  const float* __restrict__ ctx,
                  const float* __restrict__ gW1, const float* __restrict__ gb1,
                  const float* __restrict__ gW2,
                  const float* __restrict__ g1Wih, const float* __restrict__ g1Whh,
                  const float* __restrict__ g1bih, const float* __restrict__ g1bhh,
                  const float* __restrict__ g2Wih, const float* __restrict__ g2Whh,
                  const float* __restrict__ g2bih, const float* __restrict__ g2bhh,
                  const float* __restrict__ outW, const float* __restrict__ outb,
                  float* __restrict__ gout, float* __restrict__ out) {
    __shared__ float s_tn[MB];
    __shared__ float s_h1n[MB * GAM];
    __shared__ float s_h2n[MB * G2];
    __shared__ float s_hid[MB * 8];
    const int tid = threadIdx.x;

    if (tid < MB) s_tn[tid] = (t[tid] - 40.0f) / 20.0f;
    __syncthreads();

    // ---- GRU1: (b,g) over 8x64 -------------------------------------------
    for (int idx = tid; idx < MB * GAM; idx += 256) {
        const int b = idx >> 6, g = idx & 63;
        float ir = g1bih[g], iz = g1bih[GAM + g], in_ = g1bih[2 * GAM + g];
        const float* wr = g1Wih + (size_t)g            * (NN + 1);
        const float* wz = g1Wih + (size_t)(GAM + g)    * (NN + 1);
        const float* wn = g1Wih + (size_t)(2 * GAM + g)* (NN + 1);
        const float* xb = x + b * NN;
        for (int d = 0; d < NN; ++d) {
            const float xv = xb[d];
            ir += xv * wr[d]; iz += xv * wz[d]; in_ += xv * wn[d];
        }
        const float tn = s_tn[b];
        ir += tn * wr[NN]; iz += tn * wz[NN]; in_ += tn * wn[NN];

        float hr = g1bhh[g], hz = g1bhh[GAM + g], hn = g1bhh[2 * GAM + g];
        const float* vr = g1Whh + (size_t)g             * GAM;
        const float* vz = g1Whh + (size_t)(GAM + g)     * GAM;
        const float* vn = g1Whh + (size_t)(2 * GAM + g) * GAM;
        const float* hb = h1 + b * GAM;
        for (int d = 0; d < GAM; ++d) {
            const float hv = hb[d];
            hr += hv * vr[d]; hz += hv * vz[d]; hn += hv * vn[d];
        }
        const float r  = sigmf(ir + hr);
        const float z  = sigmf(iz + hz);
        const float nn = tanhf(in_ + r * hn);
        const float hnew = (1.0f - z) * nn + z * hb[g];
        s_h1n[idx] = hnew;
        out[O_H1N + idx] = hnew;
    }
    __syncthreads();

    // ---- GRU2: (b,g) over 8x49 -------------------------------------------
    for (int idx = tid; idx < MB * G2; idx += 256) {
        const int b = idx / G2, g = idx % G2;
        float ir = g2bih[g], iz = g2bih[G2 + g], in_ = g2bih[2 * G2 + g];
        const float* wr = g2Wih + (size_t)g            * GAM;
        const float* wz = g2Wih + (size_t)(G2 + g)     * GAM;
        const float* wn = g2Wih + (size_t)(2 * G2 + g) * GAM;
        const float* hb1 = s_h1n + b * GAM;
        for (int d = 0; d < GAM; ++d) {
            const float hv = hb1[d];
            ir += hv * wr[d]; iz += hv * wz[d]; in_ += hv * wn[d];
        }
        float hr = g2bhh[g], hz = g2bhh[G2 + g], hn = g2bhh[2 * G2 + g];
        const float* vr = g2Whh + (size_t)g            * G2;
        const float* vz = g2Whh + (size_t)(G2 + g)     * G2;
        const float* vn = g2Whh + (size_t)(2 * G2 + g) * G2;
        const float* hb = h2 + b * G2;
        for (int d = 0; d < G2; ++d) {
            const float hv = hb[d];
            hr += hv * vr[d]; hz += hv * vz[d]; hn += hv * vn[d];
        }
        const float r  = sigmf(ir + hr);
        const float z  = sigmf(iz + hz);
        const float nn = tanhf(in_ + r * hn);
        const float hnew = (1.0f - z) * nn + z * hb[g];
        s_h2n[idx] = hnew;
        out[O_H2N + idx] = hnew;
    }
    __syncthreads();

    // ---- hazard head ------------------------------------------------------
    if (tid < MB) {
        float acc = outb[0];
        for (int g = 0; g < G2; ++g) acc += eluf(s_h2n[tid * G2 + g]) * outW[g];
        out[O_LOGG  + tid] = acc;
        out[O_LOGDS + tid] = -__expf(acc);
    }

    // ---- g-net hidden (8 batches x 8 units) ------------------------------
    if (tid < 64) {
        const int b = tid >> 3, o = tid & 7;
        float acc = gb1[o];
        const float* w  = gW1 + (size_t)o * (NN + 1 + CTX);
        const float* xb = x + b * NN;
        for (int d = 0; d < NN; ++d) acc += xb[d] * w[d];
        acc += s_tn[b] * w[NN];
        #pragma unroll
        for (int c = 0; c < CTX; ++c) acc += ctx[b * CTX + c] * w[NN + 1 + c];
        s_hid[tid] = eluf(acc);
    }
    __syncthreads();

    // ---- g-net output -----------------------------------------------------
    for (int idx = tid; idx < MB * NN; idx += 256) {
        const int b = idx / NN, k = idx % NN;
        const float* h = s_hid + b * 8;
        const float* w = gW2 + (size_t)k * 8;
        float acc = 0.0f;
        #pragma unroll
        for (int o = 0; o < 8; ++o) acc += h[o] * w[o];
        gout[idx] = acc;
    }
}

// ---------------------------------------------------------------------------
// Kernel D: all dense GEMMs via V_WMMA_F32_16X16X4_F32 (M padded 8->16).
//   GEMM1: Wx  = x @ Csum^T / N  (+ f_out + g_out -> dx)
//   GEMM2: H   = elu(x @ sW1^T + b1)         (shares A-tiles with GEMM1)
//   GEMM3: sig = sigmoid(H @ sW2^T + b2)+1e-5
// 24 waves, one 16-wide output tile per wave.  A-operands are staged in a
// zero-padded 16x384 LDS tile so every A fetch is an unconditional
// ds_load_b64 (no EXEC save/restore around predicated loads; EXEC stays
// all-ones through the WMMA stream).  B fetches are global_load_b64.
// ---------------------------------------------------------------------------
__global__ __launch_bounds__(768)
void wmma_kernel(const float* __restrict__ x,   const float* __restrict__ csum,
                 const float* __restrict__ sW1, const float* __restrict__ sb1,
                 const float* __restrict__ sW2, const float* __restrict__ sb2,
                 const float* __restrict__ fout, const float* __restrict__ gout,
                 float* __restrict__ out) {
    const int tid  = threadIdx.x;
    const int wave = tid >> 5;           // 0..23
    const int lane = tid & 31;
    const int n0   = wave * 16;
    const int m    = lane & 15;          // A row / B column within tile
    const int kkb  = (lane < 16) ? 0 : 2;

    __shared__ __align__(16) float Xs[16 * NN];   // x padded 8->16 rows
    __shared__ __align__(16) float Hs[16 * NN];   // sigma hidden, padded rows

    // Stage zero-padded X; pre-zero the padding rows of Hs (rows 8..15 are
    // never written again, so GEMM3's A fetches need no predication).
    for (int idx = tid; idx < 16 * NN; idx += 768) {
        Xs[idx] = (idx < MB * NN) ? x[idx] : 0.0f;
        if (idx >= MB * NN) Hs[idx] = 0.0f;
    }
    __syncthreads();

    // ---- GEMM1 + GEMM2 fused over K (same A operand) ---------------------
    v8f acc1 = {0,0,0,0,0,0,0,0};
    v8f acc2 = {0,0,0,0,0,0,0,0};
    for (int k0 = 0; k0 < NN; k0 += 4) {
        const v2f a  = *(const v2f*)(Xs + m * NN + k0 + kkb);
        const v2f b1 = *(const v2f*)(csum + (size_t)(n0 + m) * NN + k0 + kkb);
        const v2f b2 = *(const v2f*)(sW1  + (size_t)(n0 + m) * NN + k0 + kkb);
        acc1 = __builtin_amdgcn_wmma_f32_16x16x4_f32(false, a, false, b1,
                                                     (short)0, acc1, false, false);
        acc2 = __builtin_amdgcn_wmma_f32_16x16x4_f32(false, a, false, b2,
                                                     (short)0, acc2, false, false);
    }
    if (lane < 16) {
        const int col = n0 + lane;
        #pragma unroll
        for (int e = 0; e < MB; ++e) {
            out[O_DX + e * NN + col] =
                acc1[e] * (1.0f / NN) + fout[e * NN + col] + gout[e * NN + col];
            Hs[e * NN + col] = eluf(acc2[e] + sb1[col]);
        }
    }
    __syncthreads();

    // ---- GEMM3: sigma -----------------------------------------------------
    v8f acc3 = {0,0,0,0,0,0,0,0};
    for (int k0 = 0; k0 < NN; k0 += 4) {
        const v2f a = *(const v2f*)(Hs + m * NN + k0 + kkb);
        const v2f b = *(const v2f*)(sW2 + (size_t)(n0 + m) * NN + k0 + kkb);
        acc3 = __builtin_amdgcn_wmma_f32_16x16x4_f32(false, a, false, b,
                                                     (short)0, acc3, false, false);
    }
    if (lane < 16) {
        const int col = n0 + lane;
        #pragma unroll
        for (int e = 0; e < MB; ++e)
            out[O_SIG + e * NN + col] = sigmf(acc3[e] + sb2[col]) + 1e-5f;
    }
}

// ---------------------------------------------------------------------------
extern "C" void kernel_launch(void* const* d_in, const int* in_sizes, int n_in,
                              void* d_out, int out_size, void* d_ws, size_t ws_size,
                              hipStream_t stream) {
    const float* x      = (const float*)d_in[0];
    const float* h1     = (const float*)d_in[1];
    const float* h2     = (const float*)d_in[2];
    const float* t      = (const float*)d_in[3];
    const float* ctx    = (const float*)d_in[4];
    const float* W      = (const float*)d_in[5];
    const float* fW1    = (const float*)d_in[6];
    const float* fb1    = (const float*)d_in[7];
    const float* fW2    = (const float*)d_in[8];
    const float* fb2    = (const float*)d_in[9];
    const float* gW1    = (const float*)d_in[10];
    const float* gb1    = (const float*)d_in[11];
    const float* gW2    = (const float*)d_in[12];
    const float* sW1    = (const float*)d_in[13];
    const float* sb1    = (const float*)d_in[14];
    const float* sW2    = (const float*)d_in[15];
    const float* sb2    = (const float*)d_in[16];
    const float* g1Wih  = (const float*)d_in[17];
    const float* g1Whh  = (const float*)d_in[18];
    const float* g1bih  = (const float*)d_in[19];
    const float* g1bhh  = (const float*)d_in[20];
    const float* g2Wih  = (const float*)d_in[21];
    const float* g2Whh  = (const float*)d_in[22];
    const float* g2bih  = (const float*)d_in[23];
    const float* g2bhh  = (const float*)d_in[24];
    const float* outW   = (const float*)d_in[25];
    const float* outb   = (const float*)d_in[26];

    float* ws_f  = (float*)d_ws;
    float* csum  = ws_f;                      // 384*384
    float* fout  = csum + NN * NN;            // 8*384
    float* gout  = fout + MB * NN;            // 8*384
    float* outp  = (float*)d_out;

    wreduce_kernel<<<NN, NN, 0, stream>>>(W, csum);
    fdiag_kernel<<<NN, 256, 0, stream>>>(x, t, ctx, fW1, fb1, fW2, fb2, fout);
    small_kernel<<<1, 256, 0, stream>>>(x, h1, h2, t, ctx, gW1, gb1, gW2,
                                        g1Wih, g1Whh, g1bih, g1bhh,
                                        g2Wih, g2Whh, g2bih, g2bhh,
                                        outW, outb, gout, outp);
    wmma_kernel<<<1, 768, 0, stream>>>(x, csum, sW1, sb1, sW2, sb2,
                                       fout, gout, outp);
}